// Conv2d_61881888800824
// MI455X (gfx1250) — compile-verified
//
#include <hip/hip_runtime.h>

typedef float v2f __attribute__((ext_vector_type(2)));
typedef float v8f __attribute__((ext_vector_type(8)));

#define CIN   512
#define COUT  256
#define HIN   64
#define HOUT  128
#define NBATCH 8

// ---------------- Kernel 1: per-cout inverse L2 norm ----------------
__global__ __launch_bounds__(256)
void wnorm_kernel(const float* __restrict__ w, float* __restrict__ inv) {
    __shared__ float red[256];
    const int o = blockIdx.x;
    const float* wo = w + (size_t)o * CIN * 9;
    float s = 0.f;
    for (int i = threadIdx.x; i < CIN * 9; i += 256) {
        float v = wo[i];
        s += v * v;
    }
    red[threadIdx.x] = s;
    __syncthreads();
    for (int off = 128; off > 0; off >>= 1) {
        if ((int)threadIdx.x < off) red[threadIdx.x] += red[threadIdx.x + off];
        __syncthreads();
    }
    if (threadIdx.x == 0) inv[o] = rsqrtf(red[0] + 1e-8f);
}

// ---------------- Kernel 2: build per-phase 3x3 kernels ----------------
// Wg layout: [c][tap(9)][phase(4)][o(256)]  (phase = py*2+px, tap = (dy+1)*3+(dx+1))
__global__ __launch_bounds__(256)
void wphase_kernel(const float* __restrict__ w, const float* __restrict__ inv,
                   float* __restrict__ wg) {
    const int t = blockIdx.x * 256 + threadIdx.x;   // 131072 total
    const int o = t & 255;
    const int c = t >> 8;
    const float iv = inv[o];
    float w9[3][3];
    const float* wp = w + ((size_t)o * CIN + c) * 9;
#pragma unroll
    for (int p = 0; p < 3; ++p)
#pragma unroll
        for (int q = 0; q < 3; ++q)
            w9[p][q] = wp[p * 3 + q] * iv;
    const float v4[4] = {1.f, 3.f, 3.f, 1.f};
#pragma unroll
    for (int py = 0; py < 2; ++py)
#pragma unroll
    for (int px = 0; px < 2; ++px) {
        const int phase = py * 2 + px;
#pragma unroll
        for (int dy = -1; dy <= 1; ++dy)
#pragma unroll
        for (int dx = -1; dx <= 1; ++dx) {
            const int a  = (py == 0 ? 3 : 2) + 2 * dy;   // FIR row index of 6x6 kernel
            const int bb = (px == 0 ? 3 : 2) + 2 * dx;
            float acc = 0.f;
#pragma unroll
            for (int p = 0; p < 3; ++p) {
                const int s = a - p;
                if (s < 0 || s > 3) continue;
#pragma unroll
                for (int q = 0; q < 3; ++q) {
                    const int u = bb - q;
                    if (u < 0 || u > 3) continue;
                    acc += w9[2 - p][2 - q] * v4[s] * v4[u];  // flip(w) conv f
                }
            }
            acc *= (1.f / 16.f);   // f = outer([1,3,3,1])/64 * up^2
            const int tap = (dy + 1) * 3 + (dx + 1);
            wg[(((size_t)c * 9 + tap) * 4 + phase) * 256 + o] = acc;
        }
    }
}

// ---------------- Kernel 3: implicit-GEMM conv via V_WMMA_F32_16X16X4_F32 ----
// Tile: 16 cout x (2 rows x 64 cols) spatial x 4 phases. 8 waves, wave n-tiles.
__global__ __launch_bounds__(256)
void conv_kernel(const float* __restrict__ x, const float* __restrict__ wg,
                 const float* __restrict__ bias, const float* __restrict__ noise,
                 const float* __restrict__ nstr, float* __restrict__ out) {
    // k-innermost LDS layouts for conflict-free b64 fragment loads
    __shared__ float xs[4 * 4 * 68 * 4];        // [kk][row(4)][col(68)][k(4)]
    __shared__ float wl[4 * 36 * 16 * 4];       // [kk][tap*4+ph][o(16)][k(4)]

    const int tid = threadIdx.x;
    const int wv = tid >> 5;
    const int lane = tid & 31;
    const int laneHalf = lane >> 4;
    const int laneN = lane & 15;

    const int coutBase = blockIdx.x * 16;
    const int stile = blockIdx.y;               // 0..255
    const int b = stile >> 5;
    const int y0 = (stile & 31) * 2;

    const int r = wv >> 2;                      // which of the 2 rows
    const int colBase = (wv & 3) * 16 + laneN;  // low-res column of this lane

    const float ns = nstr[0];

    v8f acc[4];
#pragma unroll
    for (int p = 0; p < 4; ++p)
#pragma unroll
        for (int i = 0; i < 8; ++i) acc[p][i] = 0.f;

    const float* xb = x + (size_t)b * CIN * HIN * HIN;

    for (int chunk = 0; chunk < 32; ++chunk) {
        const int c0 = chunk * 16;
        __syncthreads();
        // ---- stage x halo: 16 cin x 4 rows x 66 cols (zero-padded to 68) ----
        for (int idx = tid; idx < 16 * 4 * 68; idx += 256) {
            const int ci  = idx / (4 * 68);
            const int rem = idx - ci * (4 * 68);
            const int rr  = rem / 68;
            const int col = rem - rr * 68;
            float val = 0.f;
            const int xx = col - 1;
            const int yy = y0 - 1 + rr;
            if (col < 66 && xx >= 0 && xx < HIN && yy >= 0 && yy < HIN)
                val = xb[((c0 + ci) * HIN + yy) * HIN + xx];
            xs[(((ci >> 2) * 4 + rr) * 68 + col) * 4 + (ci & 3)] = val;
        }
        // ---- stage weights: 16 cin x 36 (tap,ph) x 16 o, float4 over o ----
        {
            const float* wgc = wg + (size_t)c0 * 36 * 256 + coutBase;
            for (int idx4 = tid; idx4 < 16 * 36 * 4; idx4 += 256) {
                const int ci  = idx4 / 144;
                const int rem = idx4 - ci * 144;
                const int tph = rem >> 2;
                const int q   = rem & 3;
                const float4 vv = *(const float4*)(wgc + ((size_t)ci * 36 + tph) * 256 + 4 * q);
                float* dst = &wl[(((ci >> 2) * 36 + tph) * 16 + 4 * q) * 4 + (ci & 3)];
                dst[0] = vv.x; dst[4] = vv.y; dst[8] = vv.z; dst[12] = vv.w;
            }
        }
        __syncthreads();

        // ---- 4 K-subchunks x 9 taps x 4 phases WMMA ----
#pragma unroll
        for (int kk = 0; kk < 4; ++kk) {
#pragma unroll
            for (int dy = -1; dy <= 1; ++dy) {
#pragma unroll
                for (int dx = -1; dx <= 1; ++dx) {
                    const int tap = (dy + 1) * 3 + (dx + 1);
                    // B fragment: K = j + 2*laneHalf, N = laneN
                    const v2f bfrag = *(const v2f*)&xs[
                        ((kk * 4 + (1 + r + dy)) * 68 + (colBase + 1 + dx)) * 4 + 2 * laneHalf];
#pragma unroll
                    for (int ph = 0; ph < 4; ++ph) {
                        // A fragment: M = laneN, K = j + 2*laneHalf
                        const v2f afrag = *(const v2f*)&wl[
                            ((kk * 36 + tap * 4 + ph) * 16 + laneN) * 4 + 2 * laneHalf];
                        acc[ph] = __builtin_amdgcn_wmma_f32_16x16x4_f32(
                            false, afrag, false, bfrag, (short)0, acc[ph], false, false);
                    }
                }
            }
        }
    }

    // ---- fused epilogue: noise + bias + lrelu*sqrt(2), pixel-shuffle store ----
    float bj[8];
#pragma unroll
    for (int j = 0; j < 8; ++j) bj[j] = bias[coutBase + j + 8 * laneHalf];

    const float SQRT2 = 1.41421356237309515f;
    const int rowLo = y0 + r;
#pragma unroll
    for (int py = 0; py < 2; ++py) {
        const int oy = 2 * rowLo + py;
        const float2 nz = *(const float2*)(noise + oy * HOUT + 2 * colBase);
        const float n0 = nz.x * ns, n1 = nz.y * ns;
        const v8f a0 = acc[py * 2 + 0];
        const v8f a1 = acc[py * 2 + 1];
#pragma unroll
        for (int j = 0; j < 8; ++j) {
            const int co = coutBase + j + 8 * laneHalf;   // D: M = j + 8*(lane/16)
            float v0 = a0[j] + n0 + bj[j];
            float v1 = a1[j] + n1 + bj[j];
            v0 = (v0 >= 0.f ? v0 : 0.2f * v0) * SQRT2;
            v1 = (v1 >= 0.f ? v1 : 0.2f * v1) * SQRT2;
            float2 st; st.x = v0; st.y = v1;
            *(float2*)(out + (((size_t)b * COUT + co) * HOUT + oy) * HOUT + 2 * colBase) = st;
        }
    }
}

extern "C" void kernel_launch(void* const* d_in, const int* in_sizes, int n_in,
                              void* d_out, int out_size, void* d_ws, size_t ws_size,
                              hipStream_t stream) {
    const float* x      = (const float*)d_in[0];
    const float* weight = (const float*)d_in[1];
    const float* bias   = (const float*)d_in[2];
    const float* noise  = (const float*)d_in[3];
    const float* nstr   = (const float*)d_in[4];
    float* out = (float*)d_out;

    float* inv = (float*)d_ws;          // 256 floats (padded to 1024)
    float* wg  = inv + 1024;            // 512*9*4*256 floats (~18.9 MB)

    wnorm_kernel<<<256, 256, 0, stream>>>(weight, inv);
    wphase_kernel<<<512, 256, 0, stream>>>(weight, inv, wg);
    dim3 grid(COUT / 16, NBATCH * (HIN / 2));   // 16 x 256
    conv_kernel<<<grid, 256, 0, stream>>>(x, wg, bias, noise, nstr, out);
}